// ListMLELoss_65317862637755
// MI455X (gfx1250) — compile-verified
//
#include <hip/hip_runtime.h>
#include <math.h>

// ---------------------------------------------------------------------------
// ListMLE loss, M=512 rows x N=8192 cols, one workgroup per row.
// CDNA5 (gfx1250): wave32, 320KB LDS/WGP, WMMA f32 16x16x4 tensor-scan,
// 3-tier bitonic sort (wave-shuffle / LDS / thread-register tiers).
// ---------------------------------------------------------------------------

typedef __attribute__((ext_vector_type(2))) float v2f;
typedef __attribute__((ext_vector_type(8))) float v8f;

#define NCOL   8192
#define NT     1024
#define PER    (NCOL / NT)      // 8 elements per thread
#define NIDX   4096             // config_idxs range [0, 4096)
#define NROW16 (NCOL / 16)      // 512 rows-of-16 for the scan carry
#define EPSF   1e-10f
#define PAD_KEY 0xBF800000u     // descending-sortable key of PAD (-1.0f)

// LDS carve-up (bytes) — dynamic LDS, ~146 KB total, 2 workgroups/WGP
#define OFF_KEYS   0
#define OFF_PREDS  (OFF_KEYS   + NCOL * 8)
#define OFF_VALS   (OFF_PREDS  + NCOL * 4)
#define OFF_FIRST  (OFF_VALS   + NCOL * 4)
#define OFF_ROWTOT (OFF_FIRST  + NIDX * 4)
#define OFF_RED    (OFF_ROWTOT + NROW16 * 4)
#define OFF_MX     (OFF_RED    + 32 * 4)
#define SMEM_BYTES (OFF_MX + 16)

typedef unsigned long long u64;

__device__ __forceinline__ unsigned desc_key(float y) {
    // monotone transform: ascending u32 order == descending float order
    unsigned u = __float_as_uint(y);
    unsigned s = (u & 0x80000000u) ? ~u : (u | 0x80000000u); // ascending
    return ~s;                                               // descending
}

__device__ __forceinline__ u64 shfl_xor64(u64 x, int mask) {
    return ((u64)(unsigned)__shfl_xor((int)(x >> 32), mask, 32) << 32) |
           (unsigned)__shfl_xor((int)(unsigned)x, mask, 32);
}

__global__ __launch_bounds__(NT) void listmle_row_kernel(
    const float* __restrict__ outputs,
    const float* __restrict__ runtime,
    const long long* __restrict__ idxs,
    float* __restrict__ row_sums)
{
    extern __shared__ char smem[];
    u64*      keys     = (u64*)(smem + OFF_KEYS);
    float*    preds    = (float*)(smem + OFF_PREDS);
    float*    vals     = (float*)(smem + OFF_VALS);
    unsigned* firstpos = (unsigned*)(smem + OFF_FIRST);
    float*    rowTot   = (float*)(smem + OFF_ROWTOT);
    float*    red      = (float*)(smem + OFF_RED);
    float*    mxsh     = (float*)(smem + OFF_MX);

    const int row = blockIdx.x;
    const int tid = threadIdx.x;
    const float*     orow = outputs + (size_t)row * NCOL;
    const float*     rrow = runtime + (size_t)row * NCOL;
    const long long* irow = idxs    + (size_t)row * NCOL;

    // ---- Phase A: dedup table + pred cache ------------------------------
    for (int i = tid; i < NIDX; i += NT) firstpos[i] = 0xFFFFFFFFu;
    __syncthreads();

    int myidx[PER];
#pragma unroll
    for (int p = 0; p < PER; ++p) {
        int j = tid + p * NT;                    // coalesced
        int v = (int)irow[j];
        myidx[p] = v;
        atomicMin(&firstpos[v], (unsigned)j);    // ds_min_u32
        preds[j] = orow[j];
        __builtin_prefetch(rrow + j, 0, 0);      // global_prefetch_b8 (used in B)
    }
    __syncthreads();

    // ---- Phase B: build sort keys (in registers), max of valid preds ----
    u64 kreg[PER];
    float lmax = -INFINITY;
#pragma unroll
    for (int p = 0; p < PER; ++p) {
        int j = tid + p * NT;
        bool dup = (firstpos[myidx[p]] != (unsigned)j);
        float yt = dup ? -1.0f : rrow[j];
        kreg[p] = ((u64)desc_key(yt) << 32) | (unsigned)j;
        if (!dup) lmax = fmaxf(lmax, preds[j]);
    }
    // wave32 + cross-wave max reduction
    for (int off = 16; off > 0; off >>= 1)
        lmax = fmaxf(lmax, __shfl_down(lmax, off, 32));
    if ((tid & 31) == 0) red[tid >> 5] = lmax;
    __syncthreads();
    if (tid < 32) {
        float v = red[tid];
        for (int off = 16; off > 0; off >>= 1)
            v = fmaxf(v, __shfl_down(v, off, 32));
        if (tid == 0) *mxsh = v;
    }
    __syncthreads();
    const float mx = *mxsh;

    // ---- Phase C: 3-tier bitonic sort of 8192 u64 keys ------------------
    // Element i is owned by thread (i & 1023) at slot p = i >> 10; lanes of a
    // wave hold consecutive i, so jj<=16 exchanges are intra-wave shuffles and
    // jj>=1024 exchanges are intra-thread register swaps.

    // C1: k = 2..32 — entirely intra-wave, register-resident (15 stages, 0 LDS)
    for (int k = 2; k <= 32; k <<= 1) {
        for (int jj = k >> 1; jj >= 1; jj >>= 1) {
#pragma unroll
            for (int p = 0; p < PER; ++p) {
                int i = tid + p * NT;
                u64 other = shfl_xor64(kreg[p], jj);
                bool up      = ((i & k) == 0);
                bool lower   = ((i & jj) == 0);
                bool keepMin = (up == lower);
                if ((kreg[p] > other) == keepMin) kreg[p] = other;
            }
        }
    }
#pragma unroll
    for (int p = 0; p < PER; ++p) keys[tid + p * NT] = kreg[p];
    __syncthreads();

    // C2: k = 64..8192 — register tier (jj>=1024) + LDS tier (512..32) +
    //                    wave-shuffle tail (16..1)
    for (int k = 64; k <= NCOL; k <<= 1) {
        // R-tier: partner in same thread
        if (k > NT) {
            u64 v[PER];
#pragma unroll
            for (int p = 0; p < PER; ++p) v[p] = keys[tid + p * NT];
            for (int jj = k >> 1; jj >= NT; jj >>= 1) {
                int d = jj >> 10;
#pragma unroll
                for (int p = 0; p < PER; ++p) {
                    int pp = p ^ d;
                    if (pp > p) {
                        int i = tid + p * NT;
                        bool up = ((i & k) == 0);
                        if ((v[p] > v[pp]) == up) { u64 t = v[p]; v[p] = v[pp]; v[pp] = t; }
                    }
                }
            }
#pragma unroll
            for (int p = 0; p < PER; ++p) keys[tid + p * NT] = v[p];
            __syncthreads();
        }

        // L-tier: cross-wave exchanges through LDS
        int jstart = (k > NT) ? (NT >> 1) : (k >> 1);
        for (int jj = jstart; jj >= 32; jj >>= 1) {
#pragma unroll
            for (int p = 0; p < PER; ++p) {
                int i   = tid + p * NT;
                int ixj = i ^ jj;
                if (ixj > i) {
                    u64 a = keys[i];
                    u64 b = keys[ixj];
                    bool up = ((i & k) == 0);
                    if ((a > b) == up) { keys[i] = b; keys[ixj] = a; }
                }
            }
            __syncthreads();
        }

        // S-tier: jj = 16..1 intra-wave shuffles, one LDS round trip
#pragma unroll
        for (int p = 0; p < PER; ++p) {
            int i = tid + p * NT;
            u64 x = keys[i];
            bool up = ((i & k) == 0);
#pragma unroll
            for (int jj = 16; jj >= 1; jj >>= 1) {
                u64 other = shfl_xor64(x, jj);
                bool lower   = ((i & jj) == 0);
                bool keepMin = (up == lower);
                if ((x > other) == keepMin) x = other;
            }
            keys[i] = x;
        }
        __syncthreads();
    }

    // ---- Phase D: exps in sorted order ----------------------------------
#pragma unroll
    for (int p = 0; p < PER; ++p) {
        int j = tid + p * NT;
        u64 k = keys[j];
        unsigned s = (unsigned)(k >> 32);
        float e = 0.0f;
        if (s != PAD_KEY) {
            int col = (int)(unsigned)k;
            e = expf(preds[col] - mx);
        }
        vals[j] = e;
    }
    __syncthreads();

    // ---- Phase E: suffix cumsum via WMMA f32 16x16x4 --------------------
    // Each wave owns one 256-element tile as a 16x16 row-major matrix X.
    // S = X * L (L lower-triangular ones) -> per-row suffix sums,
    // done as 4 accumulating K=4 WMMAs.
    {
        const int lane = tid & 31;
        const int wv   = tid >> 5;          // tile id, 0..31
        float* tile = vals + wv * 256;
        const int mA   = lane & 15;         // A-matrix row for this lane
        const int nB   = lane & 15;         // B/D column for this lane
        const int half = lane >> 4;         // 0: lanes 0-15, 1: lanes 16-31

        v8f acc = {};
#pragma unroll
        for (int kb = 0; kb < 16; kb += 4) {
            v2f a, b;
            int kA = kb + (half << 1);               // A layout: v0=K0|K2, v1=K1|K3
            a.x = tile[mA * 16 + kA];
            a.y = tile[mA * 16 + kA + 1];
            int r0 = kb + half;                      // B layout: v0=rows kb,kb+1
            int r1 = kb + 2 + half;                  //           v1=rows kb+2,kb+3
            b.x = (r0 >= nB) ? 1.0f : 0.0f;          // L[k][n] = (k >= n)
            b.y = (r1 >= nB) ? 1.0f : 0.0f;
            acc = __builtin_amdgcn_wmma_f32_16x16x4_f32(
                false, a, false, b, (short)0, acc, false, false);
        }

        // D layout: vgpr r holds D[M = half*8 + r, N = lane&15]
        const int mBase = half * 8;
#pragma unroll
        for (int r = 0; r < 8; ++r)
            tile[(mBase + r) * 16 + nB] = acc[r];
        if (nB == 0) {
#pragma unroll
            for (int r = 0; r < 8; ++r)
                rowTot[wv * 16 + mBase + r] = acc[r]; // S[m,0] = full row sum
        }
    }
    __syncthreads();

    // ---- Phase F: inclusive suffix scan of the 512 row totals -----------
    for (int off = 1; off < NROW16; off <<= 1) {
        float self = 0.f, other = 0.f;
        if (tid < NROW16) {
            self = rowTot[tid];
            if (tid + off < NROW16) other = rowTot[tid + off];
        }
        __syncthreads();
        if (tid < NROW16) rowTot[tid] = self + other;
        __syncthreads();
    }

    // ---- Phase G: obs = log(cums+eps) - (pred-mx), block-sum ------------
    float lsum = 0.0f;
#pragma unroll
    for (int p = 0; p < PER; ++p) {
        int j = tid + p * NT;
        u64 k = keys[j];
        unsigned s = (unsigned)(k >> 32);
        if (s != PAD_KEY) {
            int g = j >> 4;
            float cum = vals[j] + ((g + 1 < NROW16) ? rowTot[g + 1] : 0.0f);
            int col = (int)(unsigned)k;
            lsum += logf(cum + EPSF) - (preds[col] - mx);
        }
    }
    for (int off = 16; off > 0; off >>= 1)
        lsum += __shfl_down(lsum, off, 32);
    if ((tid & 31) == 0) red[tid >> 5] = lsum;
    __syncthreads();
    if (tid < 32) {
        float v = red[tid];
        for (int off = 16; off > 0; off >>= 1)
            v += __shfl_down(v, off, 32);
        if (tid == 0) row_sums[row] = v;
    }
}

__global__ void reduce_rows_kernel(const float* __restrict__ row_sums,
                                   float* __restrict__ out, int m)
{
    __shared__ float sh[32];
    int tid = threadIdx.x;
    float s = 0.0f;
    for (int i = tid; i < m; i += blockDim.x) s += row_sums[i];
    for (int off = 16; off > 0; off >>= 1) s += __shfl_down(s, off, 32);
    if ((tid & 31) == 0) sh[tid >> 5] = s;
    __syncthreads();
    if (tid < 32) {
        int nw = (int)(blockDim.x + 31) >> 5;
        float v = (tid < nw) ? sh[tid] : 0.0f;
        for (int off = 16; off > 0; off >>= 1) v += __shfl_down(v, off, 32);
        if (tid == 0) out[0] = v / (float)m;
    }
}

extern "C" void kernel_launch(void* const* d_in, const int* in_sizes, int n_in,
                              void* d_out, int out_size, void* d_ws, size_t ws_size,
                              hipStream_t stream) {
    const float*     outputs = (const float*)d_in[0];
    const float*     runtime = (const float*)d_in[1];
    const long long* idxs    = (const long long*)d_in[2];
    const int m = in_sizes[0] / NCOL;              // 512 rows

    float* row_sums = (float*)d_ws;                // 512 floats of scratch

    listmle_row_kernel<<<m, NT, SMEM_BYTES, stream>>>(outputs, runtime, idxs, row_sums);
    reduce_rows_kernel<<<1, 512, 0, stream>>>(row_sums, (float*)d_out, m);
}